// DynamicScatter_28948079575657
// MI455X (gfx1250) — compile-verified
//
#include <hip/hip_runtime.h>
#include <cstdint>

#define VGRID     64
#define NUM_CELLS (VGRID * VGRID * VGRID)   // 262144
#define N_POINTS  1000000
#define C_FEAT    128
#define NEG_INF   (-3.402823466e38f)

__device__ __forceinline__ int voxel_id(const int* __restrict__ coors, int p) {
    int c0 = coors[p * 3 + 0], c1 = coors[p * 3 + 1], c2 = coors[p * 3 + 2];
    return (c0 * VGRID + c1) * VGRID + c2;
}

// ---- 1) zero the per-cell counters (1 MB, L2-resident for the whole pipeline) ----
__global__ void k_zero(unsigned* __restrict__ counts) {
    int t = blockIdx.x * blockDim.x + threadIdx.x;          // NUM_CELLS/4 threads
    if (t < NUM_CELLS / 4) ((uint4*)counts)[t] = make_uint4(0u, 0u, 0u, 0u);
}

// ---- 2) histogram: count points per cell, remember each point's rank in its cell ----
__global__ void k_count(const int* __restrict__ coors,
                        unsigned* __restrict__ counts,
                        unsigned* __restrict__ rankInCell) {
    int p = blockIdx.x * blockDim.x + threadIdx.x;
    if (p >= N_POINTS) return;
    rankInCell[p] = atomicAdd(&counts[voxel_id(coors, p)], 1u);
}

// ---- 3a) per-block exclusive scan (1024 elems/block, 256 blocks); binarize=1 -> occupancy ----
__global__ void k_scan1(const unsigned* __restrict__ in,
                        unsigned* __restrict__ out,
                        unsigned* __restrict__ blockSums,
                        int binarize) {
    __shared__ unsigned s[1024];
    int t = threadIdx.x;
    int gid = blockIdx.x * 1024 + t;
    unsigned v = in[gid];
    if (binarize) v = v ? 1u : 0u;
    s[t] = v;
    __syncthreads();
    for (int off = 1; off < 1024; off <<= 1) {
        unsigned x = (t >= off) ? s[t - off] : 0u;
        __syncthreads();
        s[t] += x;
        __syncthreads();
    }
    out[gid] = s[t] - v;                                    // block-local exclusive
    if (t == 1023) blockSums[blockIdx.x] = s[t];
}

// ---- 3b) scan the 256 block sums; emit offsets and the grand total ----
__global__ void k_scan2(const unsigned* __restrict__ blockSums,
                        unsigned* __restrict__ blockOffs,
                        unsigned* __restrict__ totalPtr) {
    __shared__ unsigned s[256];
    int t = threadIdx.x;
    unsigned v = blockSums[t];
    s[t] = v;
    __syncthreads();
    for (int off = 1; off < 256; off <<= 1) {
        unsigned x = (t >= off) ? s[t - off] : 0u;
        __syncthreads();
        s[t] += x;
        __syncthreads();
    }
    blockOffs[t] = s[t] - v;
    if (t == 255) *totalPtr = s[t];
}

// ---- 3c) add block offsets ----
__global__ void k_scan3(unsigned* __restrict__ out, const unsigned* __restrict__ blockOffs) {
    int gid = blockIdx.x * blockDim.x + threadIdx.x;
    if (gid < NUM_CELLS) out[gid] += blockOffs[gid >> 10];
}

// ---- 4) counting-sort: place point index into its cell's segment ----
__global__ void k_place(const int* __restrict__ coors,
                        const unsigned* __restrict__ segStart,
                        const unsigned* __restrict__ rankInCell,
                        unsigned* __restrict__ sortedIdx) {
    int p = blockIdx.x * blockDim.x + threadIdx.x;
    if (p >= N_POINTS) return;
    int id = voxel_id(coors, p);
    sortedIdx[segStart[id] + rankInCell[p]] = (unsigned)p;
}

// ---- 5) segmented max: one wave per cell, lane covers 4 channels; no feature atomics ----
__global__ void k_reduce(const float4* __restrict__ pts4,
                         const unsigned* __restrict__ counts,
                         const unsigned* __restrict__ segStart,
                         const unsigned* __restrict__ slotRank,
                         const unsigned* __restrict__ sortedIdx,
                         const unsigned* __restrict__ Mptr,
                         float4* __restrict__ feats4) {
    int tid  = blockIdx.x * blockDim.x + threadIdx.x;       // NUM_CELLS * 32 threads
    int cell = tid >> 5;
    int lane = tid & 31;
    if (cell >= NUM_CELLS) return;
    unsigned M = *Mptr;
    if ((unsigned)cell >= M)                                // padding rows: zero (disjoint from slots < M)
        feats4[(size_t)cell * 32 + lane] = make_float4(0.f, 0.f, 0.f, 0.f);
    unsigned cnt = counts[cell];
    if (cnt == 0) return;
    unsigned s0   = segStart[cell];
    unsigned slot = slotRank[cell];
    float4 acc = make_float4(NEG_INF, NEG_INF, NEG_INF, NEG_INF);
    for (unsigned j = 0; j < cnt; ++j) {
        unsigned p = sortedIdx[s0 + j];                     // wave-uniform index
        if (lane == 0 && j + 1 < cnt)                       // keep the dependent gather ahead
            __builtin_prefetch((const void*)(pts4 + (size_t)sortedIdx[s0 + j + 1] * 32), 0, 0);
        float4 v = pts4[(size_t)p * 32 + lane];             // b128, full 512B row per wave
        acc.x = fmaxf(acc.x, v.x);
        acc.y = fmaxf(acc.y, v.y);
        acc.z = fmaxf(acc.z, v.z);
        acc.w = fmaxf(acc.w, v.w);
    }
    feats4[(size_t)slot * 32 + lane] = acc;                 // single plain store per lane
}

// ---- 6) voxel coords: decoded id for occupied slots, -1 padding ----
__global__ void k_coors(const unsigned* __restrict__ counts,
                        const unsigned* __restrict__ slotRank,
                        const unsigned* __restrict__ Mptr,
                        int* __restrict__ outc) {
    int i = blockIdx.x * blockDim.x + threadIdx.x;
    if (i >= NUM_CELLS) return;
    unsigned M = *Mptr;
    if (counts[i]) {
        unsigned s = slotRank[i];
        outc[s * 3 + 0] = i / (VGRID * VGRID);
        outc[s * 3 + 1] = (i / VGRID) % VGRID;
        outc[s * 3 + 2] = i % VGRID;
    }
    if ((unsigned)i >= M) {                                 // padding rows (disjoint from above)
        outc[i * 3 + 0] = -1;
        outc[i * 3 + 1] = -1;
        outc[i * 3 + 2] = -1;
    }
}

extern "C" void kernel_launch(void* const* d_in, const int* in_sizes, int n_in,
                              void* d_out, int out_size, void* d_ws, size_t ws_size,
                              hipStream_t stream) {
    const float4* pts4  = (const float4*)d_in[0];
    const int*    coors = (const int*)d_in[1];

    float4* feats4 = (float4*)d_out;
    int*    outc   = (int*)((float*)d_out + (size_t)NUM_CELLS * C_FEAT);

    // workspace layout (~11.1 MB)
    unsigned* counts     = (unsigned*)d_ws;                 // NUM_CELLS
    unsigned* segStart   = counts   + NUM_CELLS;            // NUM_CELLS
    unsigned* slotRank   = segStart + NUM_CELLS;            // NUM_CELLS
    unsigned* bsumA      = slotRank + NUM_CELLS;            // 256
    unsigned* boffA      = bsumA + 256;                     // 256
    unsigned* bsumB      = boffA + 256;                     // 256
    unsigned* boffB      = bsumB + 256;                     // 256
    unsigned* Mptr       = boffB + 256;                     // 1
    unsigned* dummyTot   = Mptr + 1;                        // 1
    unsigned* rankInCell = Mptr + 8;                        // N_POINTS (16B aligned)
    unsigned* sortedIdx  = rankInCell + N_POINTS;           // N_POINTS

    const int B = 256;

    k_zero  <<<(NUM_CELLS / 4 + B - 1) / B, B, 0, stream>>>(counts);
    k_count <<<(N_POINTS + B - 1) / B, B, 0, stream>>>(coors, counts, rankInCell);

    // scan counts -> segment starts
    k_scan1 <<<NUM_CELLS / 1024, 1024, 0, stream>>>(counts, segStart, bsumB, 0);
    k_scan2 <<<1, 256, 0, stream>>>(bsumB, boffB, dummyTot);
    k_scan3 <<<NUM_CELLS / B, B, 0, stream>>>(segStart, boffB);

    // scan occupancy -> slot ranks + M
    k_scan1 <<<NUM_CELLS / 1024, 1024, 0, stream>>>(counts, slotRank, bsumA, 1);
    k_scan2 <<<1, 256, 0, stream>>>(bsumA, boffA, Mptr);
    k_scan3 <<<NUM_CELLS / B, B, 0, stream>>>(slotRank, boffA);

    k_place <<<(N_POINTS + B - 1) / B, B, 0, stream>>>(coors, segStart, rankInCell, sortedIdx);
    k_reduce<<<(NUM_CELLS * 32 + B - 1) / B, B, 0, stream>>>(pts4, counts, segStart, slotRank,
                                                             sortedIdx, Mptr, feats4);
    k_coors <<<NUM_CELLS / B, B, 0, stream>>>(counts, slotRank, Mptr, outc);
}